// dep_model_64527588655666
// MI455X (gfx1250) — compile-verified
//
#include <hip/hip_runtime.h>
#include <hip/hip_bf16.h>

typedef float v2f __attribute__((ext_vector_type(2)));
typedef float v4f __attribute__((ext_vector_type(4)));
typedef float v8f __attribute__((ext_vector_type(8)));

#define NEG_INF (-__builtin_inff())

// B=8, N=128, H2=800, FD=500, P=100
// M = B*N = 1024 rows. Padded projection stride = 512.

// ---------------------------------------------------------------------------
// Kernel 1: projection vectors (8 x 500, contracted over P=100) + 3 constants
// ---------------------------------------------------------------------------
__global__ void vecs_kernel(const float* __restrict__ Wgp, const float* __restrict__ bgp,
                            const float* __restrict__ Wsp, const float* __restrict__ bsp,
                            const float* __restrict__ Wrp, const float* __restrict__ brp,
                            const float* __restrict__ wgo, const float* __restrict__ bgo,
                            const float* __restrict__ wso, const float* __restrict__ bso,
                            float* __restrict__ vecs, float* __restrict__ consts)
{
    const int v = blockIdx.x;     // 0..7
    const int f = threadIdx.x;    // 0..511
    const float* Wm; const float* wv; int stride, coff;
    if (v < 3)      { Wm = Wgp; wv = wgo; stride = 1500; coff = v * 500; }
    else if (v < 6) { Wm = Wsp; wv = wso; stride = 1500; coff = (v - 3) * 500; }
    else            { Wm = Wrp; wv = wgo; stride = 1000; coff = (v - 6) * 500; }

    float s = 0.0f;
    if (f < 500) {
        for (int p = 0; p < 100; ++p) s += wv[p] * Wm[p * stride + coff + f];
    }
    vecs[v * 512 + f] = s;   // pad columns get 0

    if (v == 0 && f >= 500 && f < 503) {
        int which = f - 500;  // 0: c_g, 1: c_s, 2: c_r
        const float* bb = (which == 0) ? bgp : (which == 1) ? bsp : brp;
        const float* ww = (which == 1) ? wso : wgo;
        float c = 0.0f;
        for (int p = 0; p < 100; ++p) c += bb[p] * ww[p];
        c += (which == 1) ? bso[0] : bgo[0];
        consts[which] = c;
    }
}

// ---------------------------------------------------------------------------
// Kernel 2: fp32 WMMA GEMM + bias + tanh for the 4 projections.
// One wave => 16 rows x 64 cols (4 WMMA tiles). Block = 4 waves.
// K-loop: batches of 4 WMMA steps (K=16), ping-pong double buffered so the
// 20 loads of the next batch are in flight while the current 16 WMMAs run.
// f32 16x16x4 lane layout (ISA 7.12.2): A/B element K = vgpr + 2*(lane>>4),
// row/col = lane&15  -> one aligned float2 load per matrix per K-step.
// ---------------------------------------------------------------------------
struct KBatch {
    v2f a[4];
    v2f b0[4], b1[4], b2[4], b3[4];
};

__device__ __forceinline__ void load_batch(KBatch& kb,
                                           const float* __restrict__ Xp,
                                           const float* __restrict__ Wp0,
                                           const float* __restrict__ Wp1,
                                           const float* __restrict__ Wp2,
                                           const float* __restrict__ Wp3,
                                           int k0)
{
#pragma unroll
    for (int s = 0; s < 4; ++s) {
        int k = k0 + 4 * s;
        kb.a[s]  = *(const v2f*)(Xp  + k);
        kb.b0[s] = *(const v2f*)(Wp0 + k);
        kb.b1[s] = *(const v2f*)(Wp1 + k);
        kb.b2[s] = *(const v2f*)(Wp2 + k);
        kb.b3[s] = *(const v2f*)(Wp3 + k);
    }
}

#define WMMA_F32(ACC, A, B) \
    ACC = __builtin_amdgcn_wmma_f32_16x16x4_f32(false, (A), false, (B), (short)0, ACC, false, false)

__global__ void __launch_bounds__(128)
gemm_tanh(const float* __restrict__ X,
          const float* __restrict__ W0, const float* __restrict__ W1,
          const float* __restrict__ W2, const float* __restrict__ W3,
          const float* __restrict__ b0, const float* __restrict__ b1,
          const float* __restrict__ b2, const float* __restrict__ b3,
          float* __restrict__ proj)
{
    const int mat  = blockIdx.z;
    const float* W    = (mat == 0) ? W0 : (mat == 1) ? W1 : (mat == 2) ? W2 : W3;
    const float* bias = (mat == 0) ? b0 : (mat == 1) ? b1 : (mat == 2) ? b2 : b3;

    const int lane = threadIdx.x;          // 0..31
    const int wave = threadIdx.y;          // 0..3
    const int lo   = lane & 15;
    const int hi   = lane >> 4;            // 0/1

    const int mrow0 = blockIdx.y * 16;                 // row tile base
    const int col0  = blockIdx.x * 256 + wave * 64;    // 64-col strip base

    // Column indices for the 4 tiles (clamp addresses; pad results unused)
    int n0 = col0 + lo, n1 = col0 + 16 + lo, n2 = col0 + 32 + lo, n3 = col0 + 48 + lo;
    int c0 = n0 < 500 ? n0 : 499;
    int c1 = n1 < 500 ? n1 : 499;
    int c2 = n2 < 500 ? n2 : 499;
    int c3 = n3 < 500 ? n3 : 499;

    const float* Xp  = X + (size_t)(mrow0 + lo) * 800 + 2 * hi;
    const float* Wp0 = W + (size_t)c0 * 800 + 2 * hi;
    const float* Wp1 = W + (size_t)c1 * 800 + 2 * hi;
    const float* Wp2 = W + (size_t)c2 * 800 + 2 * hi;
    const float* Wp3 = W + (size_t)c3 * 800 + 2 * hi;

    v8f acc0 = {}, acc1 = {}, acc2 = {}, acc3 = {};

    KBatch cur, nxt;
    load_batch(cur, Xp, Wp0, Wp1, Wp2, Wp3, 0);

    // 800 / 16 = 50 batches = 25 ping-pong pairs.
    for (int k0 = 0; k0 < 800; k0 += 32) {
        // stage A: prefetch batch k0+16, compute batch k0
        load_batch(nxt, Xp, Wp0, Wp1, Wp2, Wp3, k0 + 16);
#pragma unroll
        for (int s = 0; s < 4; ++s) {
            WMMA_F32(acc0, cur.a[s], cur.b0[s]);
            WMMA_F32(acc1, cur.a[s], cur.b1[s]);
            WMMA_F32(acc2, cur.a[s], cur.b2[s]);
            WMMA_F32(acc3, cur.a[s], cur.b3[s]);
        }
        // stage B: prefetch batch k0+32 (last pair re-reads 784: harmless),
        // compute batch k0+16
        int knext = (k0 + 32 < 800) ? (k0 + 32) : 784;
        load_batch(cur, Xp, Wp0, Wp1, Wp2, Wp3, knext);
#pragma unroll
        for (int s = 0; s < 4; ++s) {
            WMMA_F32(acc0, nxt.a[s], nxt.b0[s]);
            WMMA_F32(acc1, nxt.a[s], nxt.b1[s]);
            WMMA_F32(acc2, nxt.a[s], nxt.b2[s]);
            WMMA_F32(acc3, nxt.a[s], nxt.b3[s]);
        }
    }

    float bv0 = bias[c0], bv1 = bias[c1], bv2 = bias[c2], bv3 = bias[c3];
    float* P = proj + (size_t)mat * 1024 * 512;
#pragma unroll
    for (int r = 0; r < 8; ++r) {
        int row = mrow0 + r + 8 * hi;           // C/D layout: M = r + 8*(lane>>4)
        float* Pr = P + (size_t)row * 512;
        Pr[n0] = tanhf(acc0[r] + bv0);
        Pr[n1] = tanhf(acc1[r] + bv1);
        Pr[n2] = tanhf(acc2[r] + bv2);
        Pr[n3] = tanhf(acc3[r] + bv3);
    }
}

// ---------------------------------------------------------------------------
// Kernel 3: per-row scalars.
//   scal[d*1024+m], d: 0=gi 1=gj 2=gk 3=si 4=sj 5=sk 6=ri ; r0[b] for b<8
// ---------------------------------------------------------------------------
__global__ void dots(const float* __restrict__ proj, const float* __restrict__ vecs,
                     float* __restrict__ scal, float* __restrict__ r0out)
{
    int t = blockIdx.x * blockDim.x + threadIdx.x;
    if (t < 7 * 1024) {
        int d = t / 1024, m = t % 1024;
        int sm, vi;  // source matrix (0=head 1=mod 2=sib 3=grd), vector index
        switch (d) {
            case 0: sm = 1; vi = 2; break;   // gi = mod . vgm
            case 1: sm = 0; vi = 1; break;   // gj = head . vgh
            case 2: sm = 3; vi = 0; break;   // gk = grd . vg
            case 3: sm = 1; vi = 5; break;   // si = mod . vsm
            case 4: sm = 2; vi = 4; break;   // sj = sib . vss
            case 5: sm = 0; vi = 3; break;   // sk = head . vsh
            default: sm = 1; vi = 7; break;  // ri = mod . vri
        }
        const float* row = proj + (size_t)sm * 1024 * 512 + (size_t)m * 512;
        const float* v   = vecs + vi * 512;
        float s = 0.0f;
        for (int f = 0; f < 500; ++f) s += row[f] * v[f];
        scal[d * 1024 + m] = s;
    } else if (t < 7 * 1024 + 8) {
        int b = t - 7 * 1024;
        const float* row = proj + (size_t)(b * 128) * 512;   // head[b, 0, :]
        const float* v   = vecs + 6 * 512;                   // vr0
        float s = 0.0f;
        for (int f = 0; f < 500; ++f) s += row[f] * v[f];
        r0out[b] = s;
    }
}

// ---------------------------------------------------------------------------
// Kernel 4: build the two (8,128,128,128) score tensors. Store-bandwidth bound:
// 128 MiB streamed out via non-temporal 128-bit stores.
// t -> (b, i, k, j4): j4 = t&31, k = (t>>5)&127, i = (t>>12)&127, b = t>>19.
// ---------------------------------------------------------------------------
__global__ void __launch_bounds__(256)
build_scores(const float* __restrict__ scal, const float* __restrict__ r0,
             const float* __restrict__ consts, float* __restrict__ out)
{
    int t  = blockIdx.x * 256 + threadIdx.x;
    int j4 = t & 31;
    int k  = (t >> 5) & 127;
    int i  = (t >> 12) & 127;
    int b  = t >> 19;
    int bi = b * 128 + i;
    int bk = b * 128 + k;

    const float* gi = scal + 0 * 1024;
    const float* gj = scal + 1 * 1024;
    const float* gk = scal + 2 * 1024;
    const float* si = scal + 3 * 1024;
    const float* sj = scal + 4 * 1024;
    const float* sk = scal + 5 * 1024;
    const float* ri = scal + 6 * 1024;

    float cg = consts[0], cs = consts[1], cr = consts[2];
    float gbase = gi[bi] + gk[bk] + cg;
    float sbase = si[bi] + sk[bk] + cs;

    int jbase = j4 * 4;
    float rootv = 0.0f;
    bool do_root = (k == 0) && (jbase == 0) && (i != 0);
    if (do_root) rootv = tanhf(r0[b] + ri[bi] + cr);

    v4f go, so;
#pragma unroll
    for (int jj = 0; jj < 4; ++jj) {
        int j = jbase + jj;
        bool valid = (i != 0) && (i != j) && (j != k);
        float gv = valid ? tanhf(gbase + gj[b * 128 + j]) : NEG_INF;
        float sv = valid ? tanhf(sbase + sj[b * 128 + j]) : 0.0f;
        if (do_root && j == 0) gv = rootv;
        go[jj] = gv;
        so[jj] = sv;
    }

    size_t off = (((size_t)bi * 128 + k) * 128 + jbase);
    __builtin_nontemporal_store(go, (v4f*)(out + off));
    __builtin_nontemporal_store(so, (v4f*)(out + (size_t)8 * 128 * 128 * 128 + off));
}

// ---------------------------------------------------------------------------
extern "C" void kernel_launch(void* const* d_in, const int* in_sizes, int n_in,
                              void* d_out, int out_size, void* d_ws, size_t ws_size,
                              hipStream_t stream)
{
    (void)in_sizes; (void)n_in; (void)out_size; (void)ws_size;

    const float* hidden = (const float*)d_in[0];
    const float* Wh  = (const float*)d_in[1];  const float* bh  = (const float*)d_in[2];
    const float* Wm_ = (const float*)d_in[3];  const float* bm  = (const float*)d_in[4];
    const float* Ws_ = (const float*)d_in[5];  const float* bs  = (const float*)d_in[6];
    const float* Wg_ = (const float*)d_in[7];  const float* bg  = (const float*)d_in[8];
    const float* Wgp = (const float*)d_in[9];  const float* bgp = (const float*)d_in[10];
    const float* Wsp = (const float*)d_in[11]; const float* bsp = (const float*)d_in[12];
    const float* Wrp = (const float*)d_in[13]; const float* brp = (const float*)d_in[14];
    const float* wgo = (const float*)d_in[15]; const float* bgo = (const float*)d_in[16];
    const float* wso = (const float*)d_in[17]; const float* bso = (const float*)d_in[18];

    float* ws     = (float*)d_ws;
    float* proj   = ws;                          // 4 * 1024 * 512 floats (8 MiB)
    float* vecs   = proj + 4 * 1024 * 512;       // 8 * 512
    float* consts = vecs + 8 * 512;              // 3 (padded to 16)
    float* scal   = consts + 16;                 // 7 * 1024
    float* r0     = scal + 7 * 1024;             // 8
    float* out    = (float*)d_out;

    vecs_kernel<<<dim3(8), dim3(512), 0, stream>>>(
        Wgp, bgp, Wsp, bsp, Wrp, brp, wgo, bgo, wso, bso, vecs, consts);

    gemm_tanh<<<dim3(2, 64, 4), dim3(32, 4, 1), 0, stream>>>(
        hidden, Wh, Wm_, Ws_, Wg_, bh, bm, bs, bg, proj);

    dots<<<dim3(29), dim3(256), 0, stream>>>(proj, vecs, scal, r0);

    build_scores<<<dim3(16384), dim3(256), 0, stream>>>(scal, r0, consts, out);
}